// AdvancedMultiHeadAttention_28381143892076
// MI455X (gfx1250) — compile-verified
//
#include <hip/hip_runtime.h>
#include <hip/hip_bf16.h>

typedef float v2f __attribute__((ext_vector_type(2)));
typedef float v8f __attribute__((ext_vector_type(8)));

#define WMMA_F32(a, b, c) \
    __builtin_amdgcn_wmma_f32_16x16x4_f32(false, (a), false, (b), (short)0, (c), false, false)

// Async global->LDS copy (gfx1250): 16B per lane, no VGPR round trip.
// vdst operand = 32-bit LDS byte address, vaddr = 64-bit global VA.
__device__ __forceinline__ void async_g2l_b128(const void* lds, const void* g) {
    const unsigned int  l  = (unsigned int)(uintptr_t)lds;   // LDS offset in [31:0]
    const unsigned long long ga = (unsigned long long)(uintptr_t)g;
    asm volatile("global_load_async_to_lds_b128 %0, %1, off"
                 :: "v"(l), "v"(ga) : "memory");
}
__device__ __forceinline__ void wait_async0() {
#if __has_builtin(__builtin_amdgcn_s_wait_asynccnt)
    __builtin_amdgcn_s_wait_asynccnt(0);
#else
    asm volatile("s_wait_asynccnt 0" ::: "memory");
#endif
}
__device__ __forceinline__ void wait_ds0() {
    asm volatile("s_wait_dscnt 0" ::: "memory");
}

// Problem constants (from reference): B=32, T=8, C=1024, S=4096, H=16, hd=64
static constexpr int kB  = 32;
static constexpr int kT  = 8;
static constexpr int kC  = 1024;
static constexpr int kS  = 4096;
static constexpr int kH  = 16;
static constexpr int kHD = 64;
static constexpr int kM  = kB * kT;  // 256 rows of activations
static constexpr int kP  = 68;       // row pad: 272B = multiple of 16B for async B128

// ---------------------------------------------------------------------------
// Generic  out[m,n] = sum_k A[m,k] * W[n,k] (+ bias[n])   via WMMA f32 16x16x4
// grid = (M/16, N/64), block = 128 threads (4 waves, one 16-col tile each)
// Tiles staged with global_load_async_to_lds_b128.
// ---------------------------------------------------------------------------
__global__ __launch_bounds__(128)
void gemm_xwT(const float* __restrict__ A, const float* __restrict__ W,
              const float* __restrict__ bias, float* __restrict__ out,
              int M, int N, int K) {
    __shared__ __align__(16) float xs[16][kP];   // A tile, padded rows
    __shared__ __align__(16) float wsm[64][kP];  // W tile (rows = out cols)

    const int tid  = threadIdx.x;
    const int lane = tid & 31;
    const int wave = tid >> 5;
    const int mbase = blockIdx.x * 16;
    const int nbase = blockIdx.y * 64;

    const int q16   = lane & 15;
    const int half2 = (lane >> 4) << 1;

    v8f c = {};
    for (int k0 = 0; k0 < K; k0 += 64) {
        // Stage A tile: 16 rows x 64 cols = 256 xB128, 2 per thread (async)
        #pragma unroll
        for (int i = 0; i < 2; ++i) {
            int f = tid + i * 128;
            int row = f >> 4, c4 = (f & 15) << 2;
            async_g2l_b128(&xs[row][c4], A + (size_t)(mbase + row) * K + k0 + c4);
        }
        // Stage W tile: 64 rows x 64 cols = 1024 xB128, 8 per thread (async)
        #pragma unroll
        for (int i = 0; i < 8; ++i) {
            int f = tid + i * 128;
            int row = f >> 4, c4 = (f & 15) << 2;
            async_g2l_b128(&wsm[row][c4], W + (size_t)(nbase + row) * K + k0 + c4);
        }
        wait_async0();
        __syncthreads();

        // A-frag lane l: A[m=l%16, k=2*(l/16)+j]; B-frag: W[nbase+w*16+l%16, k]
        const int rB = (wave << 4) + q16;
        #pragma unroll
        for (int kt = 0; kt < 16; ++kt) {
            const int col = kt * 4 + half2;
            v2f a = *(const v2f*)&xs[q16][col];
            v2f b = *(const v2f*)&wsm[rB][col];
            c = WMMA_F32(a, b, c);
        }
        __syncthreads();
    }

    const int ncol = nbase + (wave << 4) + q16;
    const float bv = bias ? bias[ncol] : 0.0f;
    #pragma unroll
    for (int r = 0; r < 8; ++r) {
        const int row = mbase + r + ((lane >> 4) << 3);  // D: m = r + 8*(lane/16)
        out[(size_t)row * N + ncol] = c[r] + bv;
    }
}

// ---------------------------------------------------------------------------
// Flash-style decode attention, one block per (b, h), 4 waves x 16 keys/iter.
// Ragged cache semantics match the reference:
//   key <  cache_len[b]          : real cached key/value
//   cache_len[b] <= key < max_cl : zero key/value (score 0 kept in softmax)
//   max_cl <= key < S            : masked out (-inf)
//   then T new keys (always valid) from the K/V projections.
// Bulk K/V tiles streamed with async global->LDS B128 copies.
// ---------------------------------------------------------------------------
__global__ __launch_bounds__(128)
void attn_kernel(const float* __restrict__ qbuf, const float* __restrict__ kbuf,
                 const float* __restrict__ vbuf,
                 const float* __restrict__ cache_k, const float* __restrict__ cache_v,
                 const int* __restrict__ cache_len, float* __restrict__ abuf) {
    __shared__ __align__(16) float qs[16][kP];        // queries (rows >=8 zero)
    __shared__ __align__(16) float ks[4][16][kP];     // per-wave K tile
    __shared__ __align__(16) float vsm[4][16][kP];    // per-wave V tile
    __shared__ __align__(16) float ps[4][16][18];     // per-wave exp(scores)
    __shared__ __align__(16) float accs[4][16][64];   // per-wave output acc
    __shared__ float mlb[4][16][2];                   // per-wave (m, l)

    const int b = blockIdx.x >> 4;   // / H
    const int h = blockIdx.x & 15;   // % H
    const int tid  = threadIdx.x;
    const int lane = tid & 31;
    const int wave = tid >> 5;
    const int q16  = lane & 15;
    const int half = lane >> 4;
    const int half2 = half << 1;

    // cache_len: B == 32 == warpSize, one entry per lane; wave-reduce max
    const int cl_lane = cache_len[lane];
    const int cl_b = __shfl(cl_lane, b, 32);
    int mcl = cl_lane;
    #pragma unroll
    for (int off = 16; off > 0; off >>= 1) mcl = max(mcl, __shfl_xor(mcl, off, 32));

    // Stage Q (8 valid rows, rest zero)
    for (int idx = tid; idx < 16 * 64; idx += 128) {
        int row = idx >> 6, c0 = idx & 63;
        qs[row][c0] = (row < kT)
            ? qbuf[(size_t)(b * kT + row) * kC + h * kHD + c0] : 0.0f;
    }
    __syncthreads();

    const float NEG = -1e30f;
    float m = NEG, lsum = 0.0f;
    v8f acc0 = {}, acc1 = {}, acc2 = {}, acc3 = {};

    const float* kb_base = cache_k + (size_t)b * kS * kC + h * kHD;
    const float* vb_base = cache_v + (size_t)b * kS * kC + h * kHD;

    const int nIter = (mcl + 63) >> 6;   // 64-key tiles over the cache region
    for (int it = 0; it <= nIter; ++it) {
        const bool newkeys = (it == nIter);
        if (newkeys && wave != 0) break;   // only wave 0 handles the T new keys
        const int kb = it * 64 + wave * 16;

        // ---- stage this wave's 16x64 K and V tiles ----
        if (!newkeys && (kb + 16 <= cl_b)) {
            // fully-valid tile: async global->LDS B128 (2 rows / instruction)
            wait_ds0();   // previous tile's ds_loads must retire before overwrite
            const int row0 = half;           // lanes 0-15 -> row r, 16-31 -> r+1
            const int colf = q16 << 2;       // 4 floats per lane
            #pragma unroll
            for (int r = 0; r < 16; r += 2) {
                const int key = r + row0;
                async_g2l_b128(&ks[wave][key][colf],
                               kb_base + (size_t)(kb + key) * kC + colf);
                async_g2l_b128(&vsm[wave][key][colf],
                               vb_base + (size_t)(kb + key) * kC + colf);
            }
            wait_async0();
        } else {
            // boundary / zero-region / new-key tile: scalar conditional path
            for (int key = 0; key < 16; ++key) {
                float2 kv = make_float2(0.f, 0.f), vv = make_float2(0.f, 0.f);
                if (!newkeys) {
                    const int kg = kb + key;
                    if (kg < cl_b) {
                        kv = *(const float2*)(kb_base + (size_t)kg * kC + lane * 2);
                        vv = *(const float2*)(vb_base + (size_t)kg * kC + lane * 2);
                    }
                } else if (key < kT) {
                    const size_t r = (size_t)(b * kT + key) * kC + h * kHD + lane * 2;
                    kv = *(const float2*)(kbuf + r);
                    vv = *(const float2*)(vbuf + r);
                }
                *(float2*)&ks[wave][key][lane * 2]  = kv;
                *(float2*)&vsm[wave][key][lane * 2] = vv;
            }
        }
        if (!newkeys && it + 1 < nIter) {   // prefetch next tile
            __builtin_prefetch(kb_base + (size_t)(kb + 64) * kC + lane * 2, 0, 1);
            __builtin_prefetch(vb_base + (size_t)(kb + 64) * kC + lane * 2, 0, 1);
        }
        __builtin_amdgcn_wave_barrier();

        // ---- scores: D[key, query] = K_tile (16xhd) x Q^T, K-dim = 64 ----
        v8f sc = {};
        #pragma unroll
        for (int kt = 0; kt < 16; ++kt) {
            const int col = kt * 4 + half2;
            v2f a  = *(const v2f*)&ks[wave][q16][col];   // A[m=key,  k]
            v2f bq = *(const v2f*)&qs[q16][col];         // B[k, n=query]
            sc = WMMA_F32(a, bq, sc);
        }

        // ---- online softmax (lane covers query q16, keys r + 8*half) ----
        float s[8];
        float tm = NEG;
        #pragma unroll
        for (int r = 0; r < 8; ++r) {
            const int keyIdx = r + half * 8;
            const bool valid = newkeys ? (keyIdx < kT) : (kb + keyIdx < mcl);
            s[r] = valid ? sc[r] * 0.125f : NEG;   // 1/sqrt(64)
            tm = fmaxf(tm, s[r]);
        }
        tm = fmaxf(tm, __shfl_xor(tm, 16, 32));
        const float m_new = fmaxf(m, tm);
        const float scale = __expf(m - m_new);
        float p[8], psum = 0.0f;
        #pragma unroll
        for (int r = 0; r < 8; ++r) {
            p[r] = (s[r] > -5e29f) ? __expf(s[r] - m_new) : 0.0f;
            psum += p[r];
        }
        psum += __shfl_xor(psum, 16, 32);
        lsum = lsum * scale + psum;
        m = m_new;

        // rescale accumulators: query index of acc row r is r (+8 in hi half)
        #pragma unroll
        for (int r = 0; r < 8; ++r) {
            const float sr = __shfl(scale, r, 32);
            acc0[r] *= sr; acc1[r] *= sr; acc2[r] *= sr; acc3[r] *= sr;
        }

        // route P through LDS into A-layout: ps[query][key]
        #pragma unroll
        for (int r = 0; r < 8; ++r) ps[wave][q16][r + half * 8] = p[r];
        __builtin_amdgcn_wave_barrier();

        // ---- O += P(16x16) x V(16x64): 4 K-steps x 4 N-tiles ----
        #pragma unroll
        for (int kt = 0; kt < 4; ++kt) {
            const int kc = kt * 4 + half2;
            v2f pa = *(const v2f*)&ps[wave][q16][kc];    // A[m=query, k=key]
            v2f bv;
            bv.x = vsm[wave][kc][0 * 16 + q16]; bv.y = vsm[wave][kc + 1][0 * 16 + q16];
            acc0 = WMMA_F32(pa, bv, acc0);
            bv.x = vsm[wave][kc][1 * 16 + q16]; bv.y = vsm[wave][kc + 1][1 * 16 + q16];
            acc1 = WMMA_F32(pa, bv, acc1);
            bv.x = vsm[wave][kc][2 * 16 + q16]; bv.y = vsm[wave][kc + 1][2 * 16 + q16];
            acc2 = WMMA_F32(pa, bv, acc2);
            bv.x = vsm[wave][kc][3 * 16 + q16]; bv.y = vsm[wave][kc + 1][3 * 16 + q16];
            acc3 = WMMA_F32(pa, bv, acc3);
        }
    }

    // ---- publish per-wave partials and merge across the 4 waves ----
    #pragma unroll
    for (int r = 0; r < 8; ++r) {
        const int q = r + half * 8;   // rows >= 8 are dummy queries
        accs[wave][q][0 * 16 + q16] = acc0[r];
        accs[wave][q][1 * 16 + q16] = acc1[r];
        accs[wave][q][2 * 16 + q16] = acc2[r];
        accs[wave][q][3 * 16 + q16] = acc3[r];
    }
    if (lane < 16) { mlb[wave][q16][0] = m; mlb[wave][q16][1] = lsum; }
    __syncthreads();

    for (int idx = tid; idx < kT * kHD; idx += 128) {
        const int q = idx >> 6, d = idx & 63;
        float M2 = NEG;
        #pragma unroll
        for (int w = 0; w < 4; ++w) M2 = fmaxf(M2, mlb[w][q][0]);
        float L = 0.0f, acc = 0.0f;
        #pragma unroll
        for (int w = 0; w < 4; ++w) {
            const float e = __expf(mlb[w][q][0] - M2);
            L   += mlb[w][q][1] * e;
            acc += accs[w][q][d] * e;
        }
        abuf[(size_t)(b * kT + q) * kC + h * kHD + d] = acc / L;
    }
}

// ---------------------------------------------------------------------------
extern "C" void kernel_launch(void* const* d_in, const int* in_sizes, int n_in,
                              void* d_out, int out_size, void* d_ws, size_t ws_size,
                              hipStream_t stream) {
    const float* x        = (const float*)d_in[0];
    const float* cache_k  = (const float*)d_in[1];
    const float* cache_v  = (const float*)d_in[2];
    const int*   cache_len= (const int*)  d_in[3];
    const float* wq       = (const float*)d_in[4];
    const float* wk       = (const float*)d_in[5];
    const float* wv       = (const float*)d_in[6];
    const float* wo       = (const float*)d_in[7];
    const float* bo       = (const float*)d_in[8];
    float* out = (float*)d_out;

    float* qbuf = (float*)d_ws;               // (256, 1024)
    float* kbuf = qbuf + (size_t)kM * kC;     // (256, 1024)
    float* vbuf = kbuf + (size_t)kM * kC;     // (256, 1024)
    float* abuf = vbuf + (size_t)kM * kC;     // (256, 1024) pre-projection attn

    const dim3 g(kM / 16, kC / 64), blk(128);
    gemm_xwT<<<g, blk, 0, stream>>>(x,    wq, nullptr, qbuf, kM, kC, kC);
    gemm_xwT<<<g, blk, 0, stream>>>(x,    wk, nullptr, kbuf, kM, kC, kC);
    gemm_xwT<<<g, blk, 0, stream>>>(x,    wv, nullptr, vbuf, kM, kC, kC);
    attn_kernel<<<kB * kH, 128, 0, stream>>>(qbuf, kbuf, vbuf,
                                             cache_k, cache_v, cache_len, abuf);
    gemm_xwT<<<g, blk, 0, stream>>>(abuf, wo, bo, out, kM, kC, kC);
}